// GCNFast_2843268349986
// MI455X (gfx1250) — compile-verified
//
#include <hip/hip_runtime.h>

typedef __attribute__((ext_vector_type(2))) float v2f;
typedef __attribute__((ext_vector_type(8))) float v8f;
typedef __attribute__((ext_vector_type(4))) unsigned int v4u;
typedef __attribute__((ext_vector_type(8))) int v8i;
typedef __attribute__((ext_vector_type(4))) int v4i;

#define NC 64
#define NS 64
#define NT 64
#define DH 128
#define BSZ 16
#define M_TOT 4096        // Ns*Nt
#define K_TOT 4096        // Nc*Nt
#define M_TILE 128
#define KB 64             // one t' slab per K block
#define NITER (K_TOT / KB)
#define A_LDA 68          // 64 + TDM pad(4 dw every 64 dw): conflict-free, 16B aligned
#define B_LDB 136         // 128 + TDM pad(8 dw every 128 dw): halves bank-disjoint
#define M_LDA 68

// Build a 2D-tile D# (CDNA5 ISA) and issue TENSOR_LOAD_TO_LDS.
// tensor_dim0/1 set to 1<<20 (never OOB for our in-bounds tiles).
__device__ __forceinline__ void tdm_load_2d(unsigned lds_off, unsigned long long ga,
                                            unsigned g1w0, unsigned tile_d0,
                                            unsigned tile_d1, unsigned stride0)
{
    v4u g0;
    g0[0] = 1u;                                                 // count=1 (valid)
    g0[1] = lds_off;                                            // LDS byte address
    g0[2] = (unsigned)ga;                                       // global_addr[31:0]
    g0[3] = ((unsigned)(ga >> 32) & 0x1FFFFFFu) | (2u << 30);   // [56:32] | type=2
    v8i g1;
    g1[0] = (int)g1w0;                          // wg_mask=0 | data_size=4B | pad cfg
    g1[1] = 0;                                  // tensor_dim0[15:0] << 16  (1<<20 -> 0)
    g1[2] = (int)(1u << 4);                     // tensor_dim0[31:16]=16 | dim1[15:0]=0
    g1[3] = (int)((1u << 4) | (tile_d0 << 16)); // dim1[31:16]=16 | tile_dim0
    g1[4] = (int)tile_d1;                       // tile_dim1 | tile_dim2=0
    g1[5] = (int)stride0;                       // tensor_dim0_stride[31:0]
    g1[6] = 0;
    g1[7] = 0;
    v4i gz;  gz[0] = gz[1] = gz[2] = gz[3] = 0; // groups 2/3 unused (2D tile)
#if defined(__clang_major__) && __clang_major__ >= 23
    v8i gz8;
    #pragma unroll
    for (int i = 0; i < 8; ++i) gz8[i] = 0;
    __builtin_amdgcn_tensor_load_to_lds(g0, g1, gz, gz, gz8, 0);
#else
    __builtin_amdgcn_tensor_load_to_lds(g0, g1, gz, gz, 0);
#endif
}

__global__ __launch_bounds__(256) void gcn_masked_gemm_wmma_tdm(
    const float* __restrict__ h,     // [16,64,64,128]
    const float* __restrict__ AA,    // [4096,4096]; only top-left 64x64 block read
    const float* __restrict__ GCW,   // [4096,4096]
    const float* __restrict__ GCB,   // [4096,128]
    float* __restrict__ out)         // [16,64,64,128]
{
    __shared__ float Mk[64 * M_LDA];            // mask block, padded
    __shared__ float At[2][M_TILE * A_LDA];     // raw GCW slabs (TDM-filled, TDM-padded)
    __shared__ float Bt[2][KB * B_LDB];         // h slabs (TDM-filled, TDM-padded)

    const int tid  = threadIdx.x;
    const int b    = blockIdx.x;                // batch = N tile (N_TILE == d_h)
    const int m0   = blockIdx.y * M_TILE;
    const int wave = tid >> 5;
    const int lane = tid & 31;
    const int lm   = lane & 15;
    const int lh   = lane >> 4;

    // LDS byte offsets (generic shared ptr low 32 bits == LDS offset per aperture map)
    const unsigned offA[2] = { (unsigned)(size_t)&At[0][0], (unsigned)(size_t)&At[1][0] };
    const unsigned offB[2] = { (unsigned)(size_t)&Bt[0][0], (unsigned)(size_t)&Bt[1][0] };

    const unsigned long long gaGCW = (unsigned long long)(size_t)GCW;
    const unsigned long long gaH   = (unsigned long long)(size_t)h
                                   + (unsigned long long)b * (NC * NT * DH) * 4ull;

    // A: tile 64(k) x 128(m) rows, row stride 4096; pad 4dw/64dw -> 0x07520000
    // B: tile 128(d) x 64(c) rows, row stride 8192; pad 8dw/128dw -> 0x0F920000
    const unsigned G1A = 0x07520000u;
    const unsigned G1B = 0x0F920000u;

    // TDM ignores EXEC -> must be a *scalar* branch so only wave 0's stream issues it
    const int wv = __builtin_amdgcn_readfirstlane(wave);

    if (wv == 0) {
        tdm_load_2d(offA[0], gaGCW + (unsigned long long)m0 * K_TOT * 4ull,
                    G1A, 64u, 128u, (unsigned)K_TOT);
        tdm_load_2d(offB[0], gaH, G1B, 128u, 64u, (unsigned)(NT * DH));
    }

    // mask block: Mk[s][c] = AA[s,c] (mask repeats with period 64 in both dims)
    #pragma unroll 4
    for (int i = 0; i < 16; ++i) {
        const int idx = i * 256 + tid;
        const int r = idx >> 6, c = idx & 63;
        Mk[r * M_LDA + c] = AA[(size_t)r * K_TOT + c];
    }

    if (wv == 0) __builtin_amdgcn_s_wait_tensorcnt(0);
    __syncthreads();

    v8f acc[8];
    #pragma unroll
    for (int n = 0; n < 8; ++n) acc[n] = (v8f)0.0f;

    for (int it = 0; it < NITER; ++it) {
        const int cur = it & 1;

        if (wv == 0 && (it + 1) < NITER) {       // prefetch next slab into other buffer
            tdm_load_2d(offA[cur ^ 1],
                        gaGCW + ((unsigned long long)m0 * K_TOT
                                 + (unsigned long long)(it + 1) * KB) * 4ull,
                        G1A, 64u, 128u, (unsigned)K_TOT);
            tdm_load_2d(offB[cur ^ 1],
                        gaH + (unsigned long long)(it + 1) * DH * 4ull,  // t' = it+1
                        G1B, 128u, 64u, (unsigned)(NT * DH));
        }

        // ---- compute: 8 waves x (16 rows x 128 cols), V_WMMA_F32_16X16X4_F32 ----
        const float* Ap = &At[cur][(wave * 16 + lm) * A_LDA];
        const float* Mp = &Mk[((wave & 3) * 16 + lm) * M_LDA];   // (m0+row)%64
        const float* Bb = &Bt[cur][0];
        for (int kk = 0; kk < KB; kk += 4) {
            const int kc = kk + 2 * lh;
            v2f av;                                  // fused relu(mask * GCW)
            av.x = fmaxf(Ap[kc]     * Mp[kc],     0.0f);
            av.y = fmaxf(Ap[kc + 1] * Mp[kc + 1], 0.0f);
            const float* Bp0 = Bb + kc * B_LDB + lm;
            const float* Bp1 = Bp0 + B_LDB;
            #pragma unroll
            for (int n = 0; n < 8; ++n) {
                v2f bv;
                bv.x = Bp0[n * 16];
                bv.y = Bp1[n * 16];
                acc[n] = __builtin_amdgcn_wmma_f32_16x16x4_f32(
                    false, av, false, bv, (short)0, acc[n], false, false);
            }
        }

        if (wv == 0 && (it + 1) < NITER) __builtin_amdgcn_s_wait_tensorcnt(0);
        __syncthreads();
    }

    // ---- epilogue: bias + relu + permute [b, i=t*64+s, d] -> [b, s, t, d] ----
    const int mrow = m0 + wave * 16;
    #pragma unroll
    for (int n = 0; n < 8; ++n) {
        #pragma unroll
        for (int r = 0; r < 8; ++r) {
            const int i = mrow + r + lh * 8;     // C layout: VGPR r = rows r / r+8
            const int d = n * 16 + lm;
            float v = acc[n][r] + GCB[(size_t)i * DH + d];
            v = fmaxf(v, 0.0f);
            const int s = i & 63;
            const int t = i >> 6;
            out[(((size_t)b * NS + s) * NT + t) * DH + d] = v;
        }
    }
}

extern "C" void kernel_launch(void* const* d_in, const int* in_sizes, int n_in,
                              void* d_out, int out_size, void* d_ws, size_t ws_size,
                              hipStream_t stream) {
    const float* h   = (const float*)d_in[0];
    // d_in[1] = e (unused by the forward pass)
    const float* AA  = (const float*)d_in[2];
    const float* GCW = (const float*)d_in[3];
    const float* GCB = (const float*)d_in[4];
    float* out = (float*)d_out;

    dim3 grid(BSZ, M_TOT / M_TILE);   // (16, 32): batch fastest -> L2 reuse of GCW strip
    dim3 block(256);                  // 8 waves (wave32)
    gcn_masked_gemm_wmma_tdm<<<grid, block, 0, stream>>>(h, AA, GCW, GCB, out);
}